// Embedding_loss_49709951484027
// MI455X (gfx1250) — compile-verified
//
#include <hip/hip_runtime.h>
#include <hip/hip_bf16.h>

// ---------------------------------------------------------------------------
// Embedding loss for MI455X (gfx1250, wave32).
//   Phase 1 (bandwidth-bound, ~125 MB @ 23.3 TB/s): masked per-instance
//     moments via float4 + packed-mask u32 loads, deterministic LDS trees.
//   Phase 3 (matrix): each 16x16 tile of (mean_i - mean_j)^2 is computed
//     entirely by one V_WMMA_F32_16X16X4_F32 (rank-3 bilinear form in the
//     K=4 slots). Branch-free operand build and hinge masking.
// ---------------------------------------------------------------------------

typedef __attribute__((ext_vector_type(2))) float v2f;
typedef __attribute__((ext_vector_type(8))) float v8f;

namespace {
constexpr int kH    = 512;
constexpr int kW    = 512;
constexpr int kHW   = kH * kW;       // 262144
constexpr int kK    = 100;           // instances
constexpr int kKPad = 112;           // 7 tiles of 16
constexpr int kT1D  = kKPad / 16;    // 7
constexpr int kNBlk = 256;           // blocks per instance (phase 1)
constexpr int kThr  = 256;           // threads per block
}

// ---------------------------------------------------------------------------
// Kernel 1: per-(instance, block) partial {cnt, s1, s2}
// ---------------------------------------------------------------------------
__global__ __launch_bounds__(kThr) void moments_partial_kernel(
    const float* __restrict__ pred,          // (C,H,W) f32
    const unsigned char* __restrict__ mask,  // (K,H,W) bool as u8
    const int* __restrict__ cls,             // (K,)
    float* __restrict__ pcnt,                // (K, kNBlk)
    float* __restrict__ ps1,
    float* __restrict__ ps2)
{
    const int k   = blockIdx.y;
    const int blk = blockIdx.x;
    const int t   = threadIdx.x;
    const int c   = cls[k];

    const size_t pix = (size_t)blk * (kHW / kNBlk) + (size_t)t * 4;

    const float4 v = *(const float4*)(pred + (size_t)c * kHW + pix);
    const unsigned int mm = *(const unsigned int*)(mask + (size_t)k * kHW + pix);

    const float m0 = (float)( mm        & 0xffu);
    const float m1 = (float)((mm >> 8)  & 0xffu);
    const float m2 = (float)((mm >> 16) & 0xffu);
    const float m3 = (float)((mm >> 24) & 0xffu);

    float cnt = m0 + m1 + m2 + m3;
    float s1  = fmaf(m0, v.x, fmaf(m1, v.y, fmaf(m2, v.z, m3 * v.w)));
    float s2  = fmaf(m0 * v.x, v.x,
                fmaf(m1 * v.y, v.y,
                fmaf(m2 * v.z, v.z, m3 * v.w * v.w)));

    __shared__ float sc[kThr], sa[kThr], sb[kThr];
    sc[t] = cnt; sa[t] = s1; sb[t] = s2;
    __syncthreads();
    #pragma unroll
    for (int off = kThr / 2; off > 0; off >>= 1) {
        if (t < off) { sc[t] += sc[t + off]; sa[t] += sa[t + off]; sb[t] += sb[t + off]; }
        __syncthreads();
    }
    if (t == 0) {
        const int idx = k * kNBlk + blk;
        pcnt[idx] = sc[0]; ps1[idx] = sa[0]; ps2[idx] = sb[0];
    }
}

// ---------------------------------------------------------------------------
// Kernel 2: reduce kNBlk partials per instance -> mean, population variance
// ---------------------------------------------------------------------------
__global__ __launch_bounds__(kNBlk) void stats_kernel(
    const float* __restrict__ pcnt,
    const float* __restrict__ ps1,
    const float* __restrict__ ps2,
    float* __restrict__ means,   // (K,)
    float* __restrict__ vars)    // (K,)
{
    const int k = blockIdx.x;
    const int t = threadIdx.x;
    __shared__ float sc[kNBlk], sa[kNBlk], sb[kNBlk];
    sc[t] = pcnt[k * kNBlk + t];
    sa[t] = ps1 [k * kNBlk + t];
    sb[t] = ps2 [k * kNBlk + t];
    __syncthreads();
    #pragma unroll
    for (int off = kNBlk / 2; off > 0; off >>= 1) {
        if (t < off) { sc[t] += sc[t + off]; sa[t] += sa[t + off]; sb[t] += sb[t + off]; }
        __syncthreads();
    }
    if (t == 0) {
        const float cnt = sc[0], s1 = sa[0], s2 = sb[0];
        const bool  has = cnt > 0.0f;
        const float sfc = has ? cnt : 1.0f;
        const float mean = has ? (s1 / sfc) : 0.0f;
        const float var  = has ? (s2 / sfc - mean * mean) : 0.0f;
        means[k] = mean;
        vars[k]  = var;
    }
}

// ---------------------------------------------------------------------------
// Kernel 3: pairwise hinge. One block = 8 wave32 waves; each wave owns whole
// 16x16 tiles (wave-uniform loop), so EXEC is all-ones at every v_wmma.
//
// Per tile, one V_WMMA_F32_16X16X4_F32 produces D[m][n] = (m_i - m_j)^2:
//   A: K0 = mi^2, K1 = 1, K2 = mi, K3 = 0   (lanes 0-15 hold K0/K1 in
//   B: K0 = 1, K1 = mj^2, K2 = -2*mj, K3 = 0  v0/v1, lanes 16-31 K2/K3)
// ---------------------------------------------------------------------------
__global__ __launch_bounds__(256) void pairwise_kernel(
    const int*   __restrict__ cls,
    const float* __restrict__ means,
    const float* __restrict__ vars,
    float* __restrict__ out)
{
    __shared__ float mS[kKPad];
    __shared__ int   cS[kKPad];
    __shared__ float red[256];

    const int t    = threadIdx.x;
    const int wave = t >> 5;
    const int lane = t & 31;

    if (t < kKPad) {
        mS[t] = (t < kK) ? means[t] : 0.0f;
        cS[t] = (t < kK) ? cls[t]   : (-1 - t);   // distinct sentinels for pad
    }
    __syncthreads();

    const int  hl = lane & 15;      // half-lane index
    const bool lo = lane < 16;

    float acc = 0.0f;

    for (int tile = wave; tile < kT1D * kT1D; tile += 8) {
        const int tr = tile / kT1D;     // wave-uniform
        const int tc = tile % kT1D;
        if (tr > tc) continue;          // strictly-lower tiles contribute 0

        // Branch-free operand build (loads unconditional, halves via cndmask).
        const float mi = mS[tr * 16 + hl];
        const float mj = mS[tc * 16 + hl];
        v2f a, b;
        a.x = lo ? mi * mi : mi;        // K0 | K2
        a.y = lo ? 1.0f    : 0.0f;      // K1 | K3
        b.x = lo ? 1.0f    : -2.0f * mj;
        b.y = lo ? mj * mj : 0.0f;

        v8f cacc = {};
        const v8f d = __builtin_amdgcn_wmma_f32_16x16x4_f32(
            false, a, false, b, (short)0, cacc, false, false);

        // Classes: this lane's column class + 8 consecutive row classes
        // (two aligned ds_load_b128s), then pure-arithmetic masking.
        const int gj = tc * 16 + hl;
        const int cj = cS[gj];
        const bool jok = gj < kK;
        const int base = tr * 16 + (lo ? 0 : 8);      // 32B aligned
        const int4 c0 = *(const int4*)&cS[base];
        const int4 c1 = *(const int4*)&cS[base + 4];
        const int rc[8] = {c0.x, c0.y, c0.z, c0.w, c1.x, c1.y, c1.z, c1.w};

        #pragma unroll
        for (int e = 0; e < 8; ++e) {
            const int gi = base + e;                  // global row index
            const float hinge = fmaxf(1.0f - d[e], 0.0f);
            const float w = (jok && (gi < gj) && (rc[e] == cj)) ? 1.0f : 0.0f;
            acc = fmaf(w, hinge, acc);
        }
    }

    // intra-variance mean + regularizer mean (W_INTRA = W_REG = W_INTER = 1)
    if (t < kK) {
        const float m = mS[t];
        acc += (vars[t] + m * m) * (1.0f / (float)kK);
    }

    red[t] = acc;
    __syncthreads();
    #pragma unroll
    for (int off = 128; off > 0; off >>= 1) {
        if (t < off) red[t] += red[t + off];
        __syncthreads();
    }
    if (t == 0) out[0] = red[0];
}

// ---------------------------------------------------------------------------
// Launch
// ---------------------------------------------------------------------------
extern "C" void kernel_launch(void* const* d_in, const int* in_sizes, int n_in,
                              void* d_out, int out_size, void* d_ws, size_t ws_size,
                              hipStream_t stream) {
    (void)in_sizes; (void)n_in; (void)out_size; (void)ws_size;

    const float*         pred = (const float*)d_in[0];          // (C,H,W) f32
    const unsigned char* mask = (const unsigned char*)d_in[1];  // (K,H,W) bool->u8
    const int*           cls  = (const int*)d_in[2];            // (K,) i32
    float*               out  = (float*)d_out;                  // (1,) f32

    // Workspace layout (floats): pcnt | ps1 | ps2 | means | vars
    float* ws    = (float*)d_ws;
    float* pcnt  = ws;
    float* ps1   = pcnt + (size_t)kK * kNBlk;
    float* ps2   = ps1  + (size_t)kK * kNBlk;
    float* means = ps2  + (size_t)kK * kNBlk;
    float* vars  = means + kK;

    dim3 grid1(kNBlk, kK);
    moments_partial_kernel<<<grid1, kThr, 0, stream>>>(pred, mask, cls, pcnt, ps1, ps2);
    stats_kernel<<<kK, kNBlk, 0, stream>>>(pcnt, ps1, ps2, means, vars);
    pairwise_kernel<<<1, 256, 0, stream>>>(cls, means, vars, out);
}